// FNOSurrogate_84155589198713
// MI455X (gfx1250) — compile-verified
//
#include <hip/hip_runtime.h>

typedef __attribute__((ext_vector_type(2))) float v2f;
typedef __attribute__((ext_vector_type(8))) float v8f;

#define CCH     32
#define SLEN    128
#define NMODES  16
#define NLAYERS 4
#define EPSV    1e-5f
// padded LDS strides (bank-conflict avoidance; all 16B-friendly where vectorized)
#define XS  132   // X / XN row stride over S (multiple of 4)
#define AS  68    // A-concat [SP | CW] stride (multiple of 4)
#define FTS 36    // FT stride (multiple of 4 -> b128-aligned rows)

__device__ __forceinline__ float gelu_exact(float x) {
    return 0.5f * x * (1.0f + erff(x * 0.70710678118654752f));
}

__global__ __launch_bounds__(128) void fno_fused_kernel(
    const float* __restrict__ p,        // (B,5)
    const float* __restrict__ lift_w,   // (5,32)
    const float* __restrict__ lift_b,   // (32)
    const float* __restrict__ spec_wr,  // (L,32,32,16) = (l,i,o,k)
    const float* __restrict__ spec_wi,  // (L,32,32,16)
    const float* __restrict__ conv_w,   // (L,32,32)    = (l,o,i)
    const float* __restrict__ proj_w1,  // (32,32)
    const float* __restrict__ proj_b1,  // (32)
    const float* __restrict__ proj_w2,  // (32,7)
    const float* __restrict__ proj_b2,  // (7)
    float* __restrict__ out)            // (B,7)
{
    __shared__ __align__(16) float X [CCH * XS];   // current activation (C x S)
    __shared__ __align__(16) float XN[CCH * XS];   // GEMM2 output before norm
    __shared__ __align__(16) float AC[CCH * AS];   // [SP(32x32) | CW(32x32)]
    __shared__ __align__(16) float FT[CCH * FTS];  // DFT: cols 0..15 Re, 16..31 Im
    __shared__ float ctab [SLEN];   //  cos(2*pi*t/128)          (DFT re)
    __shared__ float nstab[SLEN];   // -sin(2*pi*t/128)          (DFT im)
    __shared__ float gctab[SLEN];   //  (2/128)*cos(2*pi*t/128)  (iDFT re basis)
    __shared__ float gstab[SLEN];   // -(2/128)*sin(2*pi*t/128)  (iDFT im basis)
    __shared__ float XM[CCH];
    __shared__ float H [CCH];

    const int t    = threadIdx.x;
    const int b    = blockIdx.x;
    const int lane = t & 31;
    const int w    = t >> 5;       // wave id 0..3
    const int hf   = lane >> 4;    // half-wave: selects K pair / M offset
    const int r    = lane & 15;

    // ---- twiddle tables (period 128), signs/scales pre-folded ----
    if (t < SLEN) {
        float th = 6.28318530717958647692f * (float)t * (1.0f / 128.0f);
        float c = cosf(th), s = sinf(th);
        ctab [t] = c;
        nstab[t] = -s;
        gctab[t] = (2.0f / 128.0f) * c;
        gstab[t] = -(2.0f / 128.0f) * s;
    }

    // ---- lift: x0[c] = lift_b[c] + sum_j p[b,j]*lift_w[j,c] ----
    if (t < CCH) {
        float a = lift_b[t];
        #pragma unroll
        for (int j = 0; j < 5; ++j) a += p[b * 5 + j] * lift_w[j * CCH + t];
        XM[t] = a;
    }
    __syncthreads();
    // broadcast over S (vectorized splat)
    for (int q = t; q < CCH * SLEN / 4; q += 128) {
        int c = q >> 5, s4 = (q & 31) * 4;
        float v = XM[c];
        float4 vv = make_float4(v, v, v, v);
        *reinterpret_cast<float4*>(&X[c * XS + s4]) = vv;
    }
    __syncthreads();

    for (int l = 0; l < NLAYERS; ++l) {
        // stage conv_w[l] (o,i) into AC cols 32..63 (conv_b cancels in instance norm)
        {
            int o  = t >> 2;
            int i0 = (t & 3) * 8;
            const float4* cw4 = reinterpret_cast<const float4*>(
                conv_w + ((size_t)l * CCH + o) * CCH + i0);
            *reinterpret_cast<float4*>(&AC[o * AS + 32 + i0])     = cw4[0];
            *reinterpret_cast<float4*>(&AC[o * AS + 32 + i0 + 4]) = cw4[1];
        }

        // ================= GEMM1: FT(32x32) = X(32x128) @ F(128x32) =================
        // F[s][j] = cos(2pi (j&15) s /128) for j<16 ; -sin(...) for j>=16
        {
            const int mi = w >> 1, ni = w & 1;
            const int m0 = mi * 16, n0 = ni * 16;
            // column j = n0 + r; mode = j & 15 == r for both tiles
            const float* tb = (n0 == 0) ? ctab : nstab;   // wave-uniform select
            v8f acc = {};
            #pragma unroll 4
            for (int ks = 0; ks < 32; ++ks) {
                int k = 4 * ks + 2 * hf;           // s index (K dim)
                v2f a, bf;
                a.x = X[(m0 + r) * XS + k];
                a.y = X[(m0 + r) * XS + k + 1];
                bf.x = tb[(r * k) & 127];
                bf.y = tb[(r * (k + 1)) & 127];
                acc = __builtin_amdgcn_wmma_f32_16x16x4_f32(
                        false, a, false, bf, (short)0, acc, false, false);
            }
            #pragma unroll
            for (int j = 0; j < 8; ++j)
                FT[(m0 + hf * 8 + j) * FTS + n0 + r] = acc[j];
        }
        __syncthreads();

        // ============ spectral mix (VALU, b128 loads): SP[o][k], SP[o][16+k] ============
        // thread -> (o = t>>2, modes k0..k0+3); weights contiguous => global_load_b128
        {
            const int o  = t >> 2;
            const int k0 = (t & 3) * 4;
            const float4* wr4 = reinterpret_cast<const float4*>(spec_wr)
                                + ((size_t)(l * CCH * CCH) + o) * 4 + (k0 >> 2);
            const float4* wi4 = reinterpret_cast<const float4*>(spec_wi)
                                + ((size_t)(l * CCH * CCH) + o) * 4 + (k0 >> 2);
            float4 re = make_float4(0.f, 0.f, 0.f, 0.f);
            float4 im = make_float4(0.f, 0.f, 0.f, 0.f);
            #pragma unroll 4
            for (int i = 0; i < CCH; ++i) {
                float4 fr = *reinterpret_cast<const float4*>(&FT[i * FTS + k0]);
                float4 fi = *reinterpret_cast<const float4*>(&FT[i * FTS + 16 + k0]);
                float4 wrv = wr4[i * (CCH * NMODES / 4)];
                float4 wiv = wi4[i * (CCH * NMODES / 4)];
                re.x += fr.x * wrv.x - fi.x * wiv.x;  im.x += fr.x * wiv.x + fi.x * wrv.x;
                re.y += fr.y * wrv.y - fi.y * wiv.y;  im.y += fr.y * wiv.y + fi.y * wrv.y;
                re.z += fr.z * wrv.z - fi.z * wiv.z;  im.z += fr.z * wiv.z + fi.z * wrv.z;
                re.w += fr.w * wrv.w - fi.w * wiv.w;  im.w += fr.w * wiv.w + fi.w * wrv.w;
            }
            *reinterpret_cast<float4*>(&AC[o * AS + k0])      = re;
            *reinterpret_cast<float4*>(&AC[o * AS + 16 + k0]) = im;
        }
        __syncthreads();

        // ====== GEMM2: XN(32x128) = [SP|CW](32x64) @ [G(32x128) ; X(32x128)] ======
        // G[kj][s]: kj<16 -> (1/128)*(kj==0?1:2cos) ; kj>=16 -> -(2/128)*sin (0 at k=0)
        for (int tile = 0; tile < 4; ++tile) {
            const int mi = tile >> 1;
            const int nt = (tile & 1) * 4 + w;    // N-tile 0..7
            const int m0 = mi * 16, n0 = nt * 16;
            const int sl = n0 + r;                // output column s (fixed per lane)
            v8f acc = {};
            // rows 0..15 of G: cosine part (k==0 row needs 1/128, not 2/128)
            #pragma unroll
            for (int ks = 0; ks < 4; ++ks) {
                int k = 4 * ks + 2 * hf;
                v2f a, bf;
                a.x = AC[(m0 + r) * AS + k];
                a.y = AC[(m0 + r) * AS + k + 1];
                float c0 = gctab[(k * sl) & 127];
                if (k == 0) c0 *= 0.5f;
                bf.x = c0;
                bf.y = gctab[((k + 1) * sl) & 127];
                acc = __builtin_amdgcn_wmma_f32_16x16x4_f32(
                        false, a, false, bf, (short)0, acc, false, false);
            }
            // rows 16..31 of G: -sin part (row 16 is exactly 0 since sin(0)=0)
            #pragma unroll
            for (int ks = 4; ks < 8; ++ks) {
                int k = 4 * ks + 2 * hf;          // 16..31
                int km = k - 16;
                v2f a, bf;
                a.x = AC[(m0 + r) * AS + k];
                a.y = AC[(m0 + r) * AS + k + 1];
                bf.x = gstab[(km * sl) & 127];
                bf.y = gstab[((km + 1) * sl) & 127];
                acc = __builtin_amdgcn_wmma_f32_16x16x4_f32(
                        false, a, false, bf, (short)0, acc, false, false);
            }
            // rows 32..63 = X (spatial conv contribution)
            #pragma unroll 4
            for (int ks = 8; ks < 16; ++ks) {
                int k = 4 * ks + 2 * hf;          // 32..63
                int kr = k - 32;
                v2f a, bf;
                a.x = AC[(m0 + r) * AS + k];
                a.y = AC[(m0 + r) * AS + k + 1];
                bf.x = X[kr * XS + sl];
                bf.y = X[(kr + 1) * XS + sl];
                acc = __builtin_amdgcn_wmma_f32_16x16x4_f32(
                        false, a, false, bf, (short)0, acc, false, false);
            }
            #pragma unroll
            for (int j = 0; j < 8; ++j)
                XN[(m0 + hf * 8 + j) * XS + n0 + r] = acc[j];
        }
        __syncthreads();

        // ============== instance norm (affine=False) + exact GELU ==============
        for (int rr = 0; rr < 8; ++rr) {
            int row = w * 8 + rr;
            float4 v = *reinterpret_cast<const float4*>(&XN[row * XS + lane * 4]);
            float s0 = v.x + v.y + v.z + v.w;
            float s1 = v.x * v.x + v.y * v.y + v.z * v.z + v.w * v.w;
            #pragma unroll
            for (int off = 16; off > 0; off >>= 1) {
                s0 += __shfl_xor(s0, off, 32);
                s1 += __shfl_xor(s1, off, 32);
            }
            float mu  = s0 * (1.0f / 128.0f);
            float var = s1 * (1.0f / 128.0f) - mu * mu;
            float rs  = rsqrtf(var + EPSV);
            float4 g;
            g.x = gelu_exact((v.x - mu) * rs);
            g.y = gelu_exact((v.y - mu) * rs);
            g.z = gelu_exact((v.z - mu) * rs);
            g.w = gelu_exact((v.w - mu) * rs);
            *reinterpret_cast<float4*>(&X[row * XS + lane * 4]) = g;
        }
        __syncthreads();
    }

    // ============================= head =============================
    if (t < CCH) {
        float4 s4 = make_float4(0.f, 0.f, 0.f, 0.f);
        #pragma unroll 4
        for (int s = 0; s < SLEN / 4; ++s) {
            float4 v = *reinterpret_cast<const float4*>(&X[t * XS + s * 4]);
            s4.x += v.x; s4.y += v.y; s4.z += v.z; s4.w += v.w;
        }
        XM[t] = (s4.x + s4.y + s4.z + s4.w) * (1.0f / 128.0f);
    }
    __syncthreads();
    if (t < CCH) {
        float a = proj_b1[t];
        #pragma unroll 4
        for (int c = 0; c < CCH; ++c) a += XM[c] * proj_w1[c * CCH + t];
        H[t] = gelu_exact(a);
    }
    __syncthreads();
    if (t < 7) {
        float a = proj_b2[t];
        #pragma unroll 4
        for (int c = 0; c < CCH; ++c) a += H[c] * proj_w2[c * 7 + t];
        out[b * 7 + t] = fmaxf(a, 0.0f);
    }
}

extern "C" void kernel_launch(void* const* d_in, const int* in_sizes, int n_in,
                              void* d_out, int out_size, void* d_ws, size_t ws_size,
                              hipStream_t stream) {
    (void)n_in; (void)out_size; (void)d_ws; (void)ws_size;
    const float* p       = (const float*)d_in[0];
    const float* lift_w  = (const float*)d_in[1];
    const float* lift_b  = (const float*)d_in[2];
    const float* spec_wr = (const float*)d_in[3];
    const float* spec_wi = (const float*)d_in[4];
    const float* conv_w  = (const float*)d_in[5];
    // d_in[6] = conv_b: constant over S -> cancels exactly inside instance norm
    const float* proj_w1 = (const float*)d_in[7];
    const float* proj_b1 = (const float*)d_in[8];
    const float* proj_w2 = (const float*)d_in[9];
    const float* proj_b2 = (const float*)d_in[10];
    float* outp = (float*)d_out;

    const int B = in_sizes[0] / 5;   // 8192
    fno_fused_kernel<<<dim3(B), dim3(128), 0, stream>>>(
        p, lift_w, lift_b, spec_wr, spec_wi, conv_w,
        proj_w1, proj_b1, proj_w2, proj_b2, outp);
}